// GradHist_33122787786882
// MI455X (gfx1250) — compile-verified
//
#include <hip/hip_runtime.h>

// Soft histogram for MI455X (gfx1250, wave32).
// Branch-free fixed-width (10-bin) sigmoid window per sample, one v_exp_f32 per
// sample with the window stepped by u *= e^5, per-wave LDS accumulators
// (ds_add_f32), and an exact f32 WMMA (v_wmma_f32_16x16x4_f32) cross-wave
// reduction: D = ones(16x4) x B(4x16) + C sums 4 wave copies per WMMA.

typedef __attribute__((ext_vector_type(2))) float v2f;
typedef __attribute__((ext_vector_type(8))) float v8f;

#define BINS      512
#define LROW      (BINS + 1)         // +1 trash slot for out-of-range bins
#define NWAVE     8                  // waves per workgroup (wave32)
#define NTHREADS  256
#define SPT       8                  // samples per thread (2x float4)
#define SPW       (NTHREADS * SPT)   // 2048 samples per workgroup
#define WIN       10                 // fixed window width in bins

#define E5   148.4131591f            // e^{+5}
#define EM5  6.7379470e-3f           // e^{-5}

template <bool ATOMIC>
__global__ __launch_bounds__(NTHREADS)
void GradHist_hist_kernel(const float* __restrict__ x, float* __restrict__ dst,
                          int S, int wgs_per_batch)
{
    __shared__ float lh[NWAVE][LROW];

    const int tid  = threadIdx.x;
    const int wave = tid >> 5;
    const int lane = tid & 31;

    // zero LDS accumulators (incl. trash slots)
    for (int i = tid; i < NWAVE * LROW; i += NTHREADS)
        (&lh[0][0])[i] = 0.0f;
    __syncthreads();

    const int b  = blockIdx.x / wgs_per_batch;
    const int wg = blockIdx.x - b * wgs_per_batch;
    const float* xb = x + (size_t)b * S + (size_t)wg * SPW;
    float* mh = lh[wave];

    // ---- windowed soft-histogram accumulation ----
    // t = x + 255.5 ; s_k = sigmoid(5*(t-k)); diffs saturate to 0.0f in fp32
    // outside a 10-bin window around t, so only that window is evaluated.
#pragma unroll
    for (int v = 0; v < SPT / 4; ++v) {
        const int si = (tid + v * NTHREADS) * 4;
        if ((size_t)wg * SPW + si + 3 < (size_t)S) {
            const float4 q = ((const float4*)xb)[tid + v * NTHREADS];
            const float vals[4] = {q.x, q.y, q.z, q.w};
#pragma unroll
            for (int e = 0; e < 4; ++e) {
                const float t = vals[e] + 255.5f;
                int kc = (int)floorf(t);
                kc = min(max(kc, -5), BINS + 4);   // window always overlaps [0,511]
                const int kbase = kc - 4;
                // u_k = exp(-5*(t-k)); start at k = kbase, step u *= e^5
                float u = __expf(-5.0f * (t - (float)kbase));
                float sprev = __builtin_amdgcn_rcpf(1.0f + u * EM5); // s_{kbase-1}
#pragma unroll
                for (int i = 0; i < WIN; ++i) {
                    const int   k  = kbase + i;
                    const float sk = __builtin_amdgcn_rcpf(1.0f + u);
                    float d = sprev - sk;
                    if (k == 0) d = 1.0f - sk;     // reference's exact top row
                    // out-of-range bins go to the per-wave trash slot (no exec churn)
                    const int kk = ((unsigned)k < (unsigned)BINS) ? k : BINS;
                    atomicAdd(&mh[kk], d);         // ds_add_f32
                    sprev = sk;
                    u *= E5;
                }
            }
        }
    }
    __syncthreads();

    // ---- cross-wave reduction via exact f32 WMMA ----
    // D(16x16) = ones(16x4) * B(4x16) + C : every row of D holds the column
    // sums of B. Two chained WMMAs sum all 8 wave copies in full f32.
    const v2f ones   = {1.0f, 1.0f};
    const int nl     = lane & 15;
    const int hiHalf = lane >> 4;
#pragma unroll
    for (int g = 0; g < 4; ++g) {                 // 8 waves x 4 groups = 32 x 16 bins
        const int col = (wave * 4 + g) * 16 + nl; // bin column handled by this lane
        v2f b0, b1;
        b0[0] = lh[hiHalf ? 2 : 0][col];
        b0[1] = lh[hiHalf ? 3 : 1][col];
        b1[0] = lh[hiHalf ? 6 : 4][col];
        b1[1] = lh[hiHalf ? 7 : 5][col];
        v8f c = {};
        c = __builtin_amdgcn_wmma_f32_16x16x4_f32(false, ones, false, b0,
                                                  (short)0, c, false, false);
        c = __builtin_amdgcn_wmma_f32_16x16x4_f32(false, ones, false, b1,
                                                  (short)0, c, false, false);
        const float sum = c[0];   // D[0][col] (lanes 0-15) == column sum of 8 copies
        if (lane < 16) {
            if (ATOMIC)
                atomicAdd(&dst[b * BINS + col], sum);
            else
                dst[(size_t)blockIdx.x * BINS + col] = sum;
        }
    }
}

__global__ void GradHist_zero_kernel(float* __restrict__ out, int n)
{
    const int i = blockIdx.x * blockDim.x + threadIdx.x;
    if (i < n) out[i] = 0.0f;
}

__global__ void GradHist_reduce_kernel(const float* __restrict__ p,
                                       float* __restrict__ out,
                                       int wgs_per_batch)
{
    const int idx = blockIdx.x * blockDim.x + threadIdx.x;  // b*BINS + k
    const int b = idx >> 9;
    const int k = idx & (BINS - 1);
    const float* pp = p + ((size_t)b * wgs_per_batch) * BINS + k;
    float acc = 0.0f;
    for (int w = 0; w < wgs_per_batch; ++w)
        acc += pp[(size_t)w * BINS];
    out[idx] = acc;
}

extern "C" void kernel_launch(void* const* d_in, const int* in_sizes, int n_in,
                              void* d_out, int out_size, void* d_ws, size_t ws_size,
                              hipStream_t stream)
{
    const float* x   = (const float*)d_in[0];
    float*       out = (float*)d_out;

    const int B = out_size / BINS;                    // 8
    const int S = in_sizes[0] / (B > 0 ? B : 1);      // 65536
    const int wgs_per_batch = (S + SPW - 1) / SPW;    // 32
    const int nwg = B * wgs_per_batch;                // 256

    const size_t need = (size_t)nwg * BINS * sizeof(float);
    if (ws_size >= need) {
        // deterministic two-pass: per-WG partials in d_ws, fixed-order reduce
        float* partial = (float*)d_ws;
        GradHist_hist_kernel<false><<<nwg, NTHREADS, 0, stream>>>(x, partial, S,
                                                                  wgs_per_batch);
        GradHist_reduce_kernel<<<(out_size + 255) / 256, 256, 0, stream>>>(
            partial, out, wgs_per_batch);
    } else {
        // fallback: zero output, accumulate with global f32 atomics
        GradHist_zero_kernel<<<(out_size + 255) / 256, 256, 0, stream>>>(out,
                                                                         out_size);
        GradHist_hist_kernel<true><<<nwg, NTHREADS, 0, stream>>>(x, out, S,
                                                                 wgs_per_batch);
    }
}